// SDN_6064493822554
// MI455X (gfx1250) — compile-verified
//
#include <hip/hip_runtime.h>
#include <stdint.h>

// SDN room-acoustics scan for MI455X (gfx1250, wave32).
// grid = 4 (one workgroup per mic), block = 32 (one wave).
// All delay-line state lives in LDS; scattering uses V_WMMA_F32_16X16X4_F32
// (8 ops/step, two independent depth-4 chains via pad-row block alignment);
// input signal streamed global->LDS with async-to-LDS double buffering.

#define NNODE 6
#define NM1   5
#define NLINE 30
#define FIRT  8          // FIR + 1 taps
#define TLEN  4000
#define BLEN  272        // ring length: max delay 261 + 8 taps < 272
#define BATCH 4
#define GGEOM (343.0 / 16000.0)

typedef float v2f __attribute__((ext_vector_type(2)));
typedef float v8f __attribute__((ext_vector_type(8)));

__device__ __forceinline__ void async_f32_to_lds(const float* g, float* l) {
  unsigned lds = (unsigned)(uintptr_t)l;                 // low 32 bits = LDS byte address
  unsigned long long ga = (unsigned long long)(uintptr_t)g;
  asm volatile("global_load_async_to_lds_b32 %0, %1, off"
               :: "v"(lds), "v"(ga) : "memory");
}

// line index -> padded row (nodes 0-2 in rows 0-14, nodes 3-5 in rows 16-30)
__device__ __forceinline__ int padrow(int l) { return l + (l >= 15 ? 1 : 0); }

// padded 32x32 view of the block-diagonal scattering matrix
__device__ __forceinline__ float sbd_pad(int r, int c,
                                         const float S[NNODE][NM1][NM1]) {
  if (r == 15 || r == 31 || c == 15 || c == 31 || r > 31 || c > 31) return 0.0f;
  int lr = r - (r > 15 ? 1 : 0);
  int lc = c - (c > 15 ? 1 : 0);
  int rn = lr / NM1, cn = lc / NM1;
  return (rn == cn) ? S[rn][lr % NM1][lc % NM1] : 0.0f;
}

__global__ __launch_bounds__(32)
void sdn_scan_kernel(const float* __restrict__ x,
                     const float* __restrict__ srcp,
                     const float* __restrict__ micp,
                     const float* __restrict__ jf,
                     const float* __restrict__ jvs,
                     const float* __restrict__ pw,
                     float* __restrict__ out)
{
  __shared__ float xs[2][32];            // async double-buffer for input chunks
  __shared__ float b1[NNODE][BLEN];      // source->node delay lines
  __shared__ float b2[NLINE][BLEN];      // node->node delay lines
  __shared__ float b3[NNODE][BLEN];      // node->mic delay lines
  __shared__ float b4[BLEN];             // direct-path delay line
  __shared__ float PPm[32][16];          // padded pressure vector (col 0 live)
  __shared__ float PMm[32][16];          // padded scattered vector
  __shared__ float DEL[32];              // FIR'd line outputs (pre-permutation)
  __shared__ float ymp[NNODE];           // per-node mic partials
  __shared__ float sS[NNODE][NM1][NM1];  // scattering blocks
  __shared__ float sUU[NNODE];
  __shared__ float sFilt[NNODE][FIRT];   // time-reversed junction filters
  __shared__ float sPw[NLINE];
  __shared__ int   sDsn[NNODE], sDnm[NNODE], sDnn[NLINE], sDsm;
  __shared__ float sSrcg[NNODE], sMicg[NNODE], sDirg;

  const int lane = threadIdx.x;
  const int b    = blockIdx.x;

  // ---- zero all scan state (fresh every launch: deterministic) ----
  for (int i = lane; i < NNODE * BLEN; i += 32) ((float*)b1)[i] = 0.0f;
  for (int i = lane; i < NLINE * BLEN; i += 32) ((float*)b2)[i] = 0.0f;
  for (int i = lane; i < NNODE * BLEN; i += 32) ((float*)b3)[i] = 0.0f;
  for (int i = lane; i < BLEN; i += 32) b4[i] = 0.0f;
  for (int i = lane; i < 512; i += 32) { ((float*)PPm)[i] = 0.0f; ((float*)PMm)[i] = 0.0f; }
  DEL[lane] = 0.0f;
  if (lane < NNODE) ymp[lane] = 0.0f;

  // ---- kick off async prefetch of input chunk 0 ----
  __builtin_prefetch(x, 0, 0);
  async_f32_to_lds(x + lane, &xs[0][lane]);

  // ---- geometry in float64 (matches numpy reference exactly) ----
  if (lane == 0) {
    const double room[3] = {4.0, 3.0, 2.5};
    double S0[3], Mp[3];
    for (int c = 0; c < 3; ++c) { S0[c] = (double)srcp[c]; Mp[c] = (double)micp[3 * b + c]; }
    double nod[NNODE][3];
    for (int j = 0; j < NNODE; ++j) {
      int a = j >> 1;
      double w = (j & 1) ? room[a] : 0.0;
      double img[3] = {S0[0], S0[1], S0[2]};
      img[a] = 2.0 * w - img[a];
      double t = (w - img[a]) / (Mp[a] - img[a]);
      for (int c = 0; c < 3; ++c) nod[j][c] = img[c] + t * (Mp[c] - img[c]);
    }
    double dsm = 0.0;
    for (int c = 0; c < 3; ++c) { double d = Mp[c] - S0[c]; dsm += d * d; }
    dsm = sqrt(dsm);
    sDsm  = (int)rint(dsm / GGEOM);
    sDirg = (float)(1.0 / dsm);
    for (int j = 0; j < NNODE; ++j) {
      double dsn = 0.0, dnm = 0.0;
      for (int c = 0; c < 3; ++c) {
        double a1 = nod[j][c] - S0[c]; dsn += a1 * a1;
        double a2 = nod[j][c] - Mp[c]; dnm += a2 * a2;
      }
      dsn = sqrt(dsn); dnm = sqrt(dnm);
      sDsn[j]  = (int)rint(dsn / GGEOM);
      sDnm[j]  = (int)rint(dnm / GGEOM);
      sSrcg[j] = (float)(0.5 * GGEOM / dsn);
      sMicg[j] = (float)(1.0 / (1.0 + dnm / dsn));
    }
    for (int p = 0; p < NLINE; ++p) {
      int i2 = p / NM1, jj = p % NM1;
      int m = jj + (jj >= i2 ? 1 : 0);               // PAIRS enumeration (m != i)
      double dnn = 0.0;
      for (int c = 0; c < 3; ++c) { double d = nod[i2][c] - nod[m][c]; dnn += d * d; }
      sDnn[p] = (int)rint(sqrt(dnn) / GGEOM);
    }
  }
  if (lane < NNODE) {
    float u = 0.0f;
    for (int j2 = 0; j2 < NM1; ++j2) { float v = jvs[lane * NM1 + j2]; u += v * v; }
    sUU[lane] = u;
  }
  if (lane < NNODE * FIRT) {
    int n = lane / FIRT, t = lane % FIRT;
    sFilt[n][t] = jf[n * FIRT + (FIRT - 1 - t)];     // time-reversed filters
  }
  if (lane < NLINE) sPw[lane] = pw[lane];
  __syncthreads();
  // scattering blocks: S[n] = 2*vs vs^T / uu - I  (float32, as in reference)
  for (int idx = lane; idx < NNODE * NM1 * NM1; idx += 32) {
    int n = idx / 25, i2 = (idx % 25) / 5, j2 = idx % 5;
    sS[n][i2][j2] = 2.0f * jvs[n * NM1 + i2] * jvs[n * NM1 + j2] / sUU[n]
                    - (i2 == j2 ? 1.0f : 0.0f);
  }
  __syncthreads();

  // ---- hoist constant A fragments of padded 32x32 S into registers ----
  // A 16x4 f32 layout: lanes 0-15 hold K=+0/+1, lanes 16-31 hold K=+2/+3.
  // Pad-row alignment makes the tiles K-disjoint: tile0 uses kc=0..3 (K 0-15),
  // tile1 uses kc=4..7 (K 16-31) -> 8 WMMAs, two independent depth-4 chains.
  v2f afr0[4], afr1[4];
  {
    const int mlo  = lane & 15;
    const int koff = (lane >> 4) << 1;
    #pragma unroll
    for (int kc = 0; kc < 4; ++kc) {
      int kb0 = kc * 4 + koff;          // K chunk for tile 0
      int kb1 = (kc + 4) * 4 + koff;    // K chunk for tile 1
      afr0[kc].x = sbd_pad(mlo,      kb0,     sS);
      afr0[kc].y = sbd_pad(mlo,      kb0 + 1, sS);
      afr1[kc].x = sbd_pad(mlo + 16, kb1,     sS);
      afr1[kc].y = sbd_pad(mlo + 16, kb1 + 1, sS);
    }
  }
  __syncthreads();

  // =======================  sequential scan  =======================
  for (int k = 0; k < TLEN; ++k) {
    // -- input chunk management: wait for current chunk, prefetch next --
    if ((k & 31) == 0) {
      asm volatile("s_wait_asynccnt 0x0" ::: "memory");
      int nk = k + 32;
      if (nk < TLEN)
        async_f32_to_lds(x + nk + lane, &xs[((k >> 5) + 1) & 1][lane]);
    }
    const float xk = xs[(k >> 5) & 1][k & 31];
    const int   km = k % BLEN;

    // -- phase 1: write source and direct-path lines (before reads) --
    if (lane < NNODE) b1[lane][km] = sSrcg[lane] * xk;
    if (lane == 0)    b4[km] = xk;
    __syncthreads();                              // cross-lane: b1 -> inj

    // -- phase 2: inject into (padded) pressure vector --
    if (lane < NLINE) {
      int n  = lane / NM1;
      int ii = k - sDsn[n];
      float inj = (ii >= 0) ? b1[n][ii % BLEN] : 0.0f;
      PPm[padrow(lane)][0] += inj;
    }
    __syncthreads();                              // cross-lane: PP -> B frags

    // -- phase 3: scattering PM = S x PP via 8 WMMA f32 ops --
    v8f acc0 = {}; v8f acc1 = {};
    {
      const int nn2  = lane & 15;
      const int koff = (lane >> 4) << 1;
      #pragma unroll
      for (int kc = 0; kc < 4; ++kc) {
        int kb0 = kc * 4 + koff;
        int kb1 = (kc + 4) * 4 + koff;
        v2f bf0, bf1;
        bf0.x = PPm[kb0][nn2];
        bf0.y = PPm[kb0 + 1][nn2];
        bf1.x = PPm[kb1][nn2];
        bf1.y = PPm[kb1 + 1][nn2];
        acc0 = __builtin_amdgcn_wmma_f32_16x16x4_f32(
            false, afr0[kc], false, bf0, (short)0, acc0, false, false);
        acc1 = __builtin_amdgcn_wmma_f32_16x16x4_f32(
            false, afr1[kc], false, bf1, (short)0, acc1, false, false);
      }
      // D layout: VGPR v -> row v (lanes 0-15) / row v+8 (lanes 16-31)
      const int rbase = (lane >> 4) << 3;
      #pragma unroll
      for (int v = 0; v < 8; ++v) {
        PMm[rbase + v][nn2]      = acc0[v];
        PMm[16 + rbase + v][nn2] = acc1[v];
      }
    }
    __syncthreads();                              // cross-lane: PM -> readers

    // -- phase 4: node pressure -> mic line; scattered vector -> lines --
    if (lane < NNODE) {
      float s = 0.0f;
      #pragma unroll
      for (int j2 = 0; j2 < NM1; ++j2)
        s += PMm[padrow(lane * NM1 + j2)][0] * sPw[lane * NM1 + j2];
      b3[lane][km] = sMicg[lane] * s;
    }
    if (lane < NLINE) b2[lane][km] = PMm[padrow(lane)][0];
    // no sync: phase-5 FIRs read b2[l]/b3[n] written by the SAME lane

    // -- phase 5: 8-tap FIRs on delayed taps --
    if (lane < NNODE) {
      float acc = 0.0f;
      #pragma unroll
      for (int t = 0; t < FIRT; ++t) {
        int ii = k - sDnm[lane] - t;
        float v = (ii >= 0) ? b3[lane][ii % BLEN] : 0.0f;
        acc += v * sFilt[lane][t];
      }
      ymp[lane] = acc;
    }
    if (lane < NLINE) {
      int n = lane / NM1;
      float acc = 0.0f;
      #pragma unroll
      for (int t = 0; t < FIRT; ++t) {
        int ii = k - sDnn[lane] - t;
        float v = (ii >= 0) ? b2[lane][ii % BLEN] : 0.0f;
        acc += v * sFilt[n][t];
      }
      DEL[lane] = acc;
    }
    __syncthreads();                              // cross-lane: DEL/ymp -> readers

    // -- phase 6: permutation and output sample --
    if (lane < NLINE) {
      int p = (6 * (lane + 1) - lane % 6 - 1) % NLINE;
      PPm[padrow(lane)][0] = DEL[p];
    }
    if (lane == 0) {
      float ym = 0.0f;
      #pragma unroll
      for (int n = 0; n < NNODE; ++n) ym += ymp[n];
      int ii = k - sDsm;
      float yd = sDirg * ((ii >= 0) ? b4[ii % BLEN] : 0.0f);
      out[b * TLEN + k] = ym + yd;
    }
    // no end-of-loop sync: next iteration's early syncs order all
    // cross-lane uses of PPm/DEL/ymp before they are touched again
  }
}

extern "C" void kernel_launch(void* const* d_in, const int* in_sizes, int n_in,
                              void* d_out, int out_size, void* d_ws, size_t ws_size,
                              hipStream_t stream) {
  const float* x    = (const float*)d_in[0];
  const float* srcp = (const float*)d_in[1];
  const float* micp = (const float*)d_in[2];
  const float* jf   = (const float*)d_in[3];
  const float* jvs  = (const float*)d_in[4];
  const float* pw   = (const float*)d_in[5];
  float* out = (float*)d_out;
  hipLaunchKernelGGL(sdn_scan_kernel, dim3(BATCH), dim3(32), 0, stream,
                     x, srcp, micp, jf, jvs, pw, out);
  (void)in_sizes; (void)n_in; (void)out_size; (void)d_ws; (void)ws_size;
}